// CachedKimiExperts_39874476376649
// MI455X (gfx1250) — compile-verified
//
#include <hip/hip_runtime.h>
#include <hip/hip_bf16.h>
#include <math.h>

// ---------------------------------------------------------------------------
// MoE (Kimi-style) top-4 of 16 experts, T=256, H=2048, I=1024, f32.
// Memory-bound: 402 MB of f32 weights stream once from HBM (~17us @ 23.3TB/s).
// Matrix math via V_WMMA_F32_16X16X4_F32 (exact f32, CDNA5 WMMA pipe).
// Each wave owns a 16-wide weight-row strip and processes TWO 16-row token
// tiles at once so every loaded B (weight) fragment feeds two WMMAs.
// ---------------------------------------------------------------------------

typedef float v2f __attribute__((ext_vector_type(2)));
typedef float v8f __attribute__((ext_vector_type(8)));

#define T_    256
#define E_    16
#define H_    2048
#define I_    1024
#define TOPK  4
#define RMAX  1536   // per-expert counts padded to 32: sum <= 1024 + 31*16 = 1520

// Workspace layout (bytes):
//   [0)      cnt[E_]          int
//   [64)     pcnt[E_]         int   (count padded to multiple of 32)
//   [128)    basev[E_]        int   (exclusive prefix of pcnt)
//   [256)    token_id[RMAX]   int
//   [6400)   gamma[RMAX]      float
//   [16384)  act[RMAX * I_]   float (~6.3 MB)
#define WS_TOKEN_OFF  256
#define WS_GAMMA_OFF  (256 + 4 * RMAX)
#define WS_ACT_OFF    16384

__device__ __forceinline__ float fast_rcp(float x) {
    return __builtin_amdgcn_rcpf(x);   // v_rcp_f32, ~1ulp
}

// ---------------------------------------------------------------------------
__global__ __launch_bounds__(256) void moe_zero_out(float* __restrict__ out, int n) {
    int i = blockIdx.x * 256 + threadIdx.x;
    if (i < n) out[i] = 0.0f;
}

// ---------------------------------------------------------------------------
// Router: softmax over 16 logits, top-4, renormalize, build per-expert slot
// lists padded to 32 rows. Single block of 256 threads (one per token).
__global__ __launch_bounds__(256) void moe_router(
    const float* __restrict__ logits,   // T_ x E_
    int* __restrict__ cnt, int* __restrict__ pcnt, int* __restrict__ basev,
    int* __restrict__ token_id, float* __restrict__ gamma)
{
    __shared__ int s_cnt[E_];
    __shared__ int s_base[E_];
    __shared__ int s_pcnt[E_];
    __shared__ int s_fill[E_];

    const int t = threadIdx.x;
    if (t < E_) { s_cnt[t] = 0; s_fill[t] = 0; }
    __syncthreads();

    // softmax numerator (denominator cancels in the top-k renormalization)
    float p[E_];
    float mx = -1e30f;
    #pragma unroll
    for (int e = 0; e < E_; ++e) { p[e] = logits[t * E_ + e]; mx = fmaxf(mx, p[e]); }
    #pragma unroll
    for (int e = 0; e < E_; ++e) { p[e] = __expf(p[e] - mx); }

    // top-4 (ties -> lowest index, matching lax.top_k)
    int   idx[TOPK];
    float val[TOPK];
    float vsum = 0.0f;
    #pragma unroll
    for (int k = 0; k < TOPK; ++k) {
        int bi = 0; float bv = -1.0f;
        #pragma unroll
        for (int e = 0; e < E_; ++e) {
            if (p[e] > bv) { bv = p[e]; bi = e; }
        }
        idx[k] = bi; val[k] = bv; vsum += bv;
        p[bi] = -2.0f;
    }

    #pragma unroll
    for (int k = 0; k < TOPK; ++k) atomicAdd(&s_cnt[idx[k]], 1);
    __syncthreads();

    if (t == 0) {
        int b = 0;
        for (int e = 0; e < E_; ++e) {
            s_base[e] = b;
            int pc = (s_cnt[e] + 31) & ~31;   // pad to 32 (two 16-row tiles)
            s_pcnt[e] = pc;
            b += pc;
        }
    }
    __syncthreads();

    const float inv = 1.0f / vsum;
    #pragma unroll
    for (int k = 0; k < TOPK; ++k) {
        int e = idx[k];
        int pos = atomicAdd(&s_fill[e], 1);
        int slot = s_base[e] + pos;
        token_id[slot] = t;
        gamma[slot] = val[k] * inv;
    }

    if (t < E_) {
        cnt[t]   = s_cnt[t];
        pcnt[t]  = s_pcnt[t];
        basev[t] = s_base[t];
        // pad tail slots: token 0, weight 0 (harmless work, zero contribution)
        for (int s = s_base[t] + s_cnt[t]; s < s_base[t] + s_pcnt[t]; ++s) {
            token_id[s] = 0;
            gamma[s] = 0.0f;
        }
    }
}

// ---------------------------------------------------------------------------
// GEMM1: act[slot, i] = silu(x_e @ w1_gate^T) * (x_e @ w1_up^T)
// One wave per (expert, 16-wide i-strip); processes m-tile PAIRS so each
// loaded weight fragment (gate + up) feeds two WMMAs.
// VGPR layouts (ISA 7.12.2):
//   A 16x4 f32 : lanes 0-15 M=lane (K=0,1), lanes 16-31 M=lane-16 (K=2,3)
//                -> per-lane float2 at column k + 2*(lane>>4)
//   B 4x16 f32 : mirrored, per-lane float2 from weight row (i0 + lane&15)
//   C/D 16x16  : lane = N, vgpr v -> M = v + 8*(lane>>4)
__global__ __launch_bounds__(32) void moe_gemm1(
    const float* __restrict__ x,        // T_ x H_
    const float* __restrict__ w1,       // E_ x (2*I_) x H_
    const int* __restrict__ pcnt, const int* __restrict__ basev,
    const int* __restrict__ token_id,
    float* __restrict__ act)            // RMAX x I_
{
    const int e    = blockIdx.x / (I_ / 16);
    const int np   = blockIdx.x % (I_ / 16);
    const int i0   = np * 16;
    const int lane = threadIdx.x;
    const int nloc = lane & 15;
    const int hi   = lane >> 4;

    const float* __restrict__ w1e = w1 + (size_t)e * (2 * I_) * H_;
    const float* __restrict__ wg  = w1e + (size_t)(i0 + nloc) * H_ + 2 * hi;        // gate row
    const float* __restrict__ wu  = w1e + (size_t)(I_ + i0 + nloc) * H_ + 2 * hi;   // up row

    const int mt2 = pcnt[e] >> 5;       // number of 32-row tile pairs
    const int b0  = basev[e];

    for (int mp = 0; mp < mt2; ++mp) {
        const int s0   = b0 + mp * 32;
        const int tok0 = token_id[s0 + nloc];
        const int tok1 = token_id[s0 + 16 + nloc];
        const float* __restrict__ xr0 = x + (size_t)tok0 * H_ + 2 * hi;
        const float* __restrict__ xr1 = x + (size_t)tok1 * H_ + 2 * hi;

        v8f c0g = {}, c0u = {}, c1g = {}, c1u = {};
        #pragma unroll 4
        for (int k = 0; k < H_; k += 4) {
            v2f a0 = *(const v2f*)(xr0 + k);
            v2f a1 = *(const v2f*)(xr1 + k);
            v2f bg = *(const v2f*)(wg + k);
            v2f bu = *(const v2f*)(wu + k);
            c0g = __builtin_amdgcn_wmma_f32_16x16x4_f32(false, a0, false, bg,
                                                        (short)0, c0g, false, false);
            c0u = __builtin_amdgcn_wmma_f32_16x16x4_f32(false, a0, false, bu,
                                                        (short)0, c0u, false, false);
            c1g = __builtin_amdgcn_wmma_f32_16x16x4_f32(false, a1, false, bg,
                                                        (short)0, c1g, false, false);
            c1u = __builtin_amdgcn_wmma_f32_16x16x4_f32(false, a1, false, bu,
                                                        (short)0, c1u, false, false);
        }

        // fused SiLU(gate) * up, store both tiles
        #pragma unroll
        for (int v = 0; v < 8; ++v) {
            const int Mrow = v + 8 * hi;
            {
                const float g = c0g[v], u = c0u[v];
                const float a = g * fast_rcp(1.0f + __expf(-g)) * u;
                act[(size_t)(s0 + Mrow) * I_ + i0 + nloc] = a;
            }
            {
                const float g = c1g[v], u = c1u[v];
                const float a = g * fast_rcp(1.0f + __expf(-g)) * u;
                act[(size_t)(s0 + 16 + Mrow) * I_ + i0 + nloc] = a;
            }
        }
    }
}

// ---------------------------------------------------------------------------
// GEMM2: out[token] += gamma[slot] * (act[slot] @ w2_e^T)
// One wave per (expert, 16-wide h-strip); m-tile pairs; f32 atomics to out.
__global__ __launch_bounds__(32) void moe_gemm2(
    const float* __restrict__ act,      // RMAX x I_
    const float* __restrict__ w2,       // E_ x H_ x I_
    const int* __restrict__ pcnt, const int* __restrict__ basev,
    const int* __restrict__ token_id, const float* __restrict__ gamma,
    float* __restrict__ out)            // T_ x H_
{
    const int e    = blockIdx.x / (H_ / 16);
    const int nh   = blockIdx.x % (H_ / 16);
    const int h0   = nh * 16;
    const int lane = threadIdx.x;
    const int nloc = lane & 15;
    const int hi   = lane >> 4;

    const float* __restrict__ w2r = w2 + (size_t)e * H_ * I_
                                       + (size_t)(h0 + nloc) * I_ + 2 * hi;

    const int mt2 = pcnt[e] >> 5;
    const int b0  = basev[e];

    for (int mp = 0; mp < mt2; ++mp) {
        const int s0 = b0 + mp * 32;
        const float* __restrict__ ar0 = act + (size_t)(s0 + nloc) * I_ + 2 * hi;
        const float* __restrict__ ar1 = act + (size_t)(s0 + 16 + nloc) * I_ + 2 * hi;

        v8f c0 = {}, c1 = {};
        #pragma unroll 4
        for (int k = 0; k < I_; k += 4) {
            v2f a0 = *(const v2f*)(ar0 + k);
            v2f a1 = *(const v2f*)(ar1 + k);
            v2f b  = *(const v2f*)(w2r + k);
            c0 = __builtin_amdgcn_wmma_f32_16x16x4_f32(false, a0, false, b,
                                                       (short)0, c0, false, false);
            c1 = __builtin_amdgcn_wmma_f32_16x16x4_f32(false, a1, false, b,
                                                       (short)0, c1, false, false);
        }

        #pragma unroll
        for (int v = 0; v < 8; ++v) {
            const int Mrow = v + 8 * hi;
            {
                const int slotM = s0 + Mrow;
                atomicAdd(&out[(size_t)token_id[slotM] * H_ + h0 + nloc],
                          gamma[slotM] * c0[v]);
            }
            {
                const int slotM = s0 + 16 + Mrow;
                atomicAdd(&out[(size_t)token_id[slotM] * H_ + h0 + nloc],
                          gamma[slotM] * c1[v]);
            }
        }
    }
}

// ---------------------------------------------------------------------------
extern "C" void kernel_launch(void* const* d_in, const int* in_sizes, int n_in,
                              void* d_out, int out_size, void* d_ws, size_t ws_size,
                              hipStream_t stream) {
    (void)in_sizes; (void)n_in; (void)ws_size; (void)out_size;

    const float* x      = (const float*)d_in[0];   // T_ x H_
    const float* logits = (const float*)d_in[1];   // T_ x E_
    const float* w1     = (const float*)d_in[2];   // E_ x 2I_ x H_
    const float* w2     = (const float*)d_in[3];   // E_ x H_ x I_
    // d_in[4] = top_k (scalar, fixed at 4)

    float* out = (float*)d_out;

    char* ws = (char*)d_ws;
    int*   cnt      = (int*)(ws + 0);
    int*   pcnt     = (int*)(ws + 64);
    int*   basev    = (int*)(ws + 128);
    int*   token_id = (int*)(ws + WS_TOKEN_OFF);
    float* gamma    = (float*)(ws + WS_GAMMA_OFF);
    float* act      = (float*)(ws + WS_ACT_OFF);

    hipLaunchKernelGGL(moe_zero_out, dim3((T_ * H_ + 255) / 256), dim3(256), 0, stream,
                       out, T_ * H_);
    hipLaunchKernelGGL(moe_router, dim3(1), dim3(256), 0, stream,
                       logits, cnt, pcnt, basev, token_id, gamma);
    hipLaunchKernelGGL(moe_gemm1, dim3(E_ * (I_ / 16)), dim3(32), 0, stream,
                       x, w1, pcnt, basev, token_id, act);
    hipLaunchKernelGGL(moe_gemm2, dim3(E_ * (H_ / 16)), dim3(32), 0, stream,
                       act, w2, pcnt, basev, token_id, gamma, out);
}